// PWRSWtL_12025908428860
// MI455X (gfx1250) — compile-verified
//
#include <hip/hip_runtime.h>
#include <hip/hip_bf16.h>

#define NBINS 256
#define LAMBDA_L2 1.0f
#define EPSV 1e-12f

typedef __attribute__((ext_vector_type(2))) float v2f;
typedef __attribute__((ext_vector_type(8))) float v8f;

// ---------------------------------------------------------------------------
// Zero the 512-float global bin array (fallback path only).
// ---------------------------------------------------------------------------
__global__ void pw_zero_bins_kernel(float* __restrict__ bins) {
    int i = threadIdx.x;
    if (i < 2 * NBINS) bins[i] = 0.0f;
}

// ---------------------------------------------------------------------------
// Fused streaming pass: per-wave LDS histograms of (count, sum of diff^2).
// mode 0: atomically flush into global bins[512]
// mode 1: write per-block partials[block][512] (deterministic reduce later)
// ---------------------------------------------------------------------------
__device__ __forceinline__ void pw_accum(float s, float t, float* wc) {
    int idx = (int)t;
    idx = idx < 0 ? 0 : (idx > (NBINS - 1) ? (NBINS - 1) : idx);
    float d = s - t;
    atomicAdd(&wc[idx], 1.0f);          // ds_add_f32
    atomicAdd(&wc[NBINS + idx], d * d); // ds_add_f32
}

__global__ __launch_bounds__(256) void pw_hist_sq_kernel(
    const float* __restrict__ src, const float* __restrict__ tar,
    float* __restrict__ out_bins, long long n, int mode) {
    __shared__ float lh[8 * 2 * NBINS]; // 8 waves x (256 counts + 256 sums) = 16 KB
    const int tid = threadIdx.x;
    float* wc = &lh[(tid >> 5) * (2 * NBINS)];

    for (int i = tid; i < 8 * 2 * NBINS; i += 256) lh[i] = 0.0f;
    __syncthreads();

    const long long n4 = n >> 2;
    const long long stride = (long long)gridDim.x * 256;
    const float4* s4 = (const float4*)src;
    const float4* t4 = (const float4*)tar;
    for (long long i = (long long)blockIdx.x * 256 + tid; i < n4; i += stride) {
        float4 s = s4[i]; // global_load_b128
        float4 t = t4[i]; // global_load_b128
        pw_accum(s.x, t.x, wc);
        pw_accum(s.y, t.y, wc);
        pw_accum(s.z, t.z, wc);
        pw_accum(s.w, t.w, wc);
    }
    // scalar tail (n % 4), handled by block 0
    if (blockIdx.x == 0 && tid < (int)(n & 3)) {
        long long i = (n4 << 2) + tid;
        pw_accum(src[i], tar[i], wc);
    }
    __syncthreads();

    // Reduce the 8 per-wave copies and flush.
    for (int b = tid; b < 2 * NBINS; b += 256) {
        float v = 0.0f;
#pragma unroll
        for (int w = 0; w < 8; ++w) v += lh[w * 2 * NBINS + b];
        if (mode == 0) {
            atomicAdd(&out_bins[b], v); // global_atomic_add_f32
        } else {
            out_bins[(long long)blockIdx.x * (2 * NBINS) + b] = v;
        }
    }
}

// ---------------------------------------------------------------------------
// Deterministic column-sum of per-block partials -> bins[512].
// Lane-consecutive bins => fully coalesced 128B reads per wave.
// ---------------------------------------------------------------------------
__global__ void pw_reduce_partials_kernel(const float* __restrict__ partials,
                                          float* __restrict__ bins, int nblocks) {
    int b = threadIdx.x + blockIdx.x * blockDim.x;
    if (b >= 2 * NBINS) return;
    float v = 0.0f;
    for (int blk = 0; blk < nblocks; ++blk)
        v += partials[(long long)blk * (2 * NBINS) + b];
    bins[b] = v;
}

// ---------------------------------------------------------------------------
// Finalize: w[b] = 1/(count[b]*inv_denom + eps); loss = L * (w.S)/(w.1).
// Both 256-element dot products are closed with V_WMMA_F32_16X16X4_F32:
//   A[m][k] = w[64c + 4m + k], B[k][n] = S[64c + 4n + k]
//   => D[m][m] += sum_k w[64c+4m+k] * S[64c+4m+k]   (diagonal accumulates dot)
// A 16x4 f32 layout (ISA 7.12.2): lane L holds M=L%16; VGPR0 = K=(L<16?0:2),
// VGPR1 = K+1. B 4x16 mirrors this => identical per-lane gather indices.
// ---------------------------------------------------------------------------
__device__ __forceinline__ float pw_v8_get(v8f v, int i) {
    switch (i & 7) {
        case 0: return v[0]; case 1: return v[1]; case 2: return v[2]; case 3: return v[3];
        case 4: return v[4]; case 5: return v[5]; case 6: return v[6]; default: return v[7];
    }
}

__global__ void pw_finalize_kernel(const float* __restrict__ bins,
                                   float* __restrict__ out, float inv_denom) {
    const int lane = threadIdx.x;       // 32 threads = one wave, EXEC all ones
    const int m = lane & 15;
    const int koff = (lane >> 4) << 1;  // 0 for lanes 0-15, 2 for lanes 16-31
    v8f c_dot = {0.f, 0.f, 0.f, 0.f, 0.f, 0.f, 0.f, 0.f};
    v8f c_sum = {0.f, 0.f, 0.f, 0.f, 0.f, 0.f, 0.f, 0.f};
    v2f ones; ones[0] = 1.0f; ones[1] = 1.0f;

#pragma unroll
    for (int ch = 0; ch < 4; ++ch) {
        int i0 = ch * 64 + 4 * m + koff;
        float c0 = bins[i0], c1 = bins[i0 + 1];
        float s0 = bins[NBINS + i0], s1 = bins[NBINS + i0 + 1];
        v2f w; w[0] = 1.0f / (c0 * inv_denom + EPSV);
               w[1] = 1.0f / (c1 * inv_denom + EPSV);
        v2f s; s[0] = s0; s[1] = s1;
        c_dot = __builtin_amdgcn_wmma_f32_16x16x4_f32(
            false, w, false, s,    (short)0, c_dot, false, false);
        c_sum = __builtin_amdgcn_wmma_f32_16x16x4_f32(
            false, w, false, ones, (short)0, c_sum, false, false);
    }

    // Diagonal extraction per C/D layout: m<8 -> VGPR m, lane m;
    // m>=8 -> VGPR m-8, lane m+16 (lanes 24..31).
    float dd = 0.0f, ws = 0.0f;
    if (lane < 8)        { dd = pw_v8_get(c_dot, lane);      ws = pw_v8_get(c_sum, lane); }
    else if (lane >= 24) { dd = pw_v8_get(c_dot, lane - 24); ws = pw_v8_get(c_sum, lane - 24); }

#pragma unroll
    for (int off = 16; off > 0; off >>= 1) {
        dd += __shfl_xor(dd, off, 32);
        ws += __shfl_xor(ws, off, 32);
    }
    if (lane == 0) out[0] = LAMBDA_L2 * dd / ws;
}

// ---------------------------------------------------------------------------
extern "C" void kernel_launch(void* const* d_in, const int* in_sizes, int n_in,
                              void* d_out, int out_size, void* d_ws, size_t ws_size,
                              hipStream_t stream) {
    const float* src = (const float*)d_in[0];
    const float* tar = (const float*)d_in[1];
    float* out = (float*)d_out;

    long long n = (long long)in_sizes[0];
    // Reference module hardcodes C=1, H=192, W=84 => per-image elems = 16128.
    long long chw = 192LL * 84LL;
    long long batch = n / chw; if (batch < 1) batch = 1;
    // p_y = counts / (numel * batch)  (faithful extra batch factor)
    float inv_denom = (float)(1.0 / ((double)n * (double)batch));

    const int MAIN_BLOCKS = 1024;
    float* ws = (float*)d_ws;
    size_t need = ((size_t)MAIN_BLOCKS * 2 * NBINS + 2 * NBINS) * sizeof(float);

    if (ws_size >= need) {
        // Deterministic path: per-block partials + tree reduce (no fp atomics).
        float* partials = ws;
        float* bins = ws + (size_t)MAIN_BLOCKS * 2 * NBINS;
        pw_hist_sq_kernel<<<MAIN_BLOCKS, 256, 0, stream>>>(src, tar, partials, n, 1);
        pw_reduce_partials_kernel<<<2, 256, 0, stream>>>(partials, bins, MAIN_BLOCKS);
        pw_finalize_kernel<<<1, 32, 0, stream>>>(bins, out, inv_denom);
    } else {
        // Fallback: tiny workspace, global atomic flush.
        float* bins = ws;
        pw_zero_bins_kernel<<<1, 512, 0, stream>>>(bins);
        pw_hist_sq_kernel<<<MAIN_BLOCKS, 256, 0, stream>>>(src, tar, bins, n, 0);
        pw_finalize_kernel<<<1, 32, 0, stream>>>(bins, out, inv_denom);
    }
}